// LIAMNE_53231824667040
// MI455X (gfx1250) — compile-verified
//
#include <hip/hip_runtime.h>

typedef __attribute__((ext_vector_type(2))) float v2f;
typedef __attribute__((ext_vector_type(8))) float v8f;

#define BATCH   4096
#define LAYERS  4
#define NEIGH   32
#define FEAT    256
#define EMB     128
#define TB      16      // batch rows per block
#define THREADS 256     // 8 wave32s
#define WAVES   8

__launch_bounds__(THREADS)
__global__ void liamne_fused(const int*   __restrict__ g_layers,
                             const int*   __restrict__ g_node_i,
                             const int*   __restrict__ g_neighs,   // [B][L][NEIGH]
                             const float* __restrict__ g_features, // [NODE][FEAT]
                             const float* __restrict__ g_layemb,   // [NODE][L][EMB]
                             const float* __restrict__ g_net,      // [L][FEAT][EMB]
                             const float* __restrict__ g_tw,       // [EMB][EMB]
                             const float* __restrict__ g_ts1,      // [EMB][EMB]
                             const float* __restrict__ g_ts2,      // [EMB]
                             float*       __restrict__ g_out)      // [B][EMB]
{
    // LDS: S (16384 f) reused later as T / AGG / FIN ; H (8192 f) ; scores/att
    __shared__ __align__(16) float S[LAYERS * TB * FEAT];   // 64 KB
    __shared__ __align__(16) float Hs[LAYERS * TB * EMB];   // 32 KB
    __shared__ float SC[LAYERS * TB];
    __shared__ float ATT[LAYERS * TB];

    const int tid  = threadIdx.x;
    const int lane = tid & 31;
    const int wave = tid >> 5;
    const int blk  = blockIdx.x;
    const int half = lane >> 4;   // WMMA K-half select
    const int lm   = lane & 15;   // WMMA row/col within half

    // ---------------- Phase A: gather + neighbor sum -> S[l][b][0..255] ----
    for (int rr = 0; rr < 8; ++rr) {
        const int row = wave * 8 + rr;           // 0..63  == l*16 + b
        const int l  = row >> 4;
        const int b  = row & 15;
        const int bg = blk * TB + b;
        const int nbase = (bg * LAYERS + l) * NEIGH;
        const int myidx = g_neighs[nbase + lane];     // one neighbor idx per lane
        float4 a0 = {0.f,0.f,0.f,0.f}, a1 = {0.f,0.f,0.f,0.f};
        for (int j = 0; j < NEIGH; ++j) {
            const int idx = __shfl(myidx, j);         // broadcast -> v_readlane
            const float4* fr = (const float4*)(g_features + (long)idx * FEAT);
            const float4 x0 = fr[lane];               // coalesced 512B per wave
            const float4 x1 = fr[32 + lane];
            a0.x += x0.x; a0.y += x0.y; a0.z += x0.z; a0.w += x0.w;
            a1.x += x1.x; a1.y += x1.y; a1.z += x1.z; a1.w += x1.w;
        }
        float4* srow = (float4*)(S + row * FEAT);
        srow[lane]      = a0;
        srow[32 + lane] = a1;
    }
    __syncthreads();

    // ---------------- Phase B: H[l] = S[l](16x256) @ net[l](256x128) -------
    // V_WMMA_F32_16X16X4_F32 : A 16x4 (v0=K0|K2, v1=K1|K3 in lane halves)
    for (int t = 0; t < 4; ++t) {
        const int tt = wave * 4 + t;                 // 0..31
        const int l  = tt >> 3;
        const int n0 = (tt & 7) * 16;
        const float* Wl   = g_net + l * FEAT * EMB;
        const float* srow = S + (l * TB + lm) * FEAT;
        v8f c = {0.f,0.f,0.f,0.f,0.f,0.f,0.f,0.f};
        for (int k = 0; k < FEAT; k += 4) {
            v2f a, bm;
            a.x  = srow[k + half*2 + 0];
            a.y  = srow[k + half*2 + 1];
            bm.x = Wl[(k + half*2 + 0) * EMB + n0 + lm];
            bm.y = Wl[(k + half*2 + 1) * EMB + n0 + lm];
            c = __builtin_amdgcn_wmma_f32_16x16x4_f32(false, a, false, bm,
                                                      (short)0, c, false, false);
        }
#pragma unroll
        for (int r = 0; r < 8; ++r) {
            const int m = half ? (r + 8) : r;        // C/D layout
            Hs[(l * TB + m) * EMB + n0 + lm] = c[r];
        }
    }
    __syncthreads();

    // ---------------- Phase C: T = tanh(H(64x128) @ Ws1(128x128)) ----------
    float* T = S;                                    // reuse S region
    for (int t = 0; t < 4; ++t) {
        const int tt = wave * 4 + t;
        const int mt = tt >> 3;                      // row tile 0..3
        const int n0 = (tt & 7) * 16;
        const float* hrow = Hs + (mt * TB + lm) * EMB;
        v8f c = {0.f,0.f,0.f,0.f,0.f,0.f,0.f,0.f};
        for (int k = 0; k < EMB; k += 4) {
            v2f a, bm;
            a.x  = hrow[k + half*2 + 0];
            a.y  = hrow[k + half*2 + 1];
            bm.x = g_ts1[(k + half*2 + 0) * EMB + n0 + lm];
            bm.y = g_ts1[(k + half*2 + 1) * EMB + n0 + lm];
            c = __builtin_amdgcn_wmma_f32_16x16x4_f32(false, a, false, bm,
                                                      (short)0, c, false, false);
        }
#pragma unroll
        for (int r = 0; r < 8; ++r) {
            const int m = half ? (r + 8) : r;
            T[(mt * TB + m) * EMB + n0 + lm] = tanhf(c[r]);
        }
    }
    __syncthreads();

    // scores[row] = sum_m T[row][m] * Ws2[m]   (wave32 reduction)
    for (int rr = 0; rr < 8; ++rr) {
        const int row = wave * 8 + rr;
        float s = 0.f;
#pragma unroll
        for (int i = 0; i < 4; ++i) {
            const int m = lane + i * 32;
            s += T[row * EMB + m] * g_ts2[m];
        }
#pragma unroll
        for (int off = 16; off > 0; off >>= 1) s += __shfl_xor(s, off);
        if (lane == 0) SC[row] = s;
    }
    __syncthreads();

    // softmax over the 4 layers, one batch row per thread
    if (tid < TB) {
        const float s0 = SC[0*TB + tid], s1 = SC[1*TB + tid];
        const float s2 = SC[2*TB + tid], s3 = SC[3*TB + tid];
        const float mx = fmaxf(fmaxf(s0, s1), fmaxf(s2, s3));
        const float e0 = expf(s0 - mx), e1 = expf(s1 - mx);
        const float e2 = expf(s2 - mx), e3 = expf(s3 - mx);
        const float inv = 1.f / (e0 + e1 + e2 + e3);
        ATT[0*TB + tid] = e0 * inv;  ATT[1*TB + tid] = e1 * inv;
        ATT[2*TB + tid] = e2 * inv;  ATT[3*TB + tid] = e3 * inv;
    }
    __syncthreads();

    // ---------------- Phase D: agg = att-weighted H ; NE = agg @ TW --------
    float* AGG = S + 8192;                 // [16][128]
    float* FIN = S + 8192 + TB * EMB;      // [16][128]
    for (int e = tid; e < TB * EMB; e += THREADS) {
        const int b = e >> 7, m = e & 127;
        float v = 0.f;
#pragma unroll
        for (int l = 0; l < LAYERS; ++l)
            v += ATT[l * TB + b] * Hs[(l * TB + b) * EMB + m];
        AGG[e] = v;
    }
    __syncthreads();

    {
        const int n0 = wave * 16;          // one 16x16 output tile per wave
        const float* arow = AGG + lm * EMB;
        v8f c = {0.f,0.f,0.f,0.f,0.f,0.f,0.f,0.f};
        for (int k = 0; k < EMB; k += 4) {
            v2f a, bm;
            a.x  = arow[k + half*2 + 0];
            a.y  = arow[k + half*2 + 1];
            bm.x = g_tw[(k + half*2 + 0) * EMB + n0 + lm];
            bm.y = g_tw[(k + half*2 + 1) * EMB + n0 + lm];
            c = __builtin_amdgcn_wmma_f32_16x16x4_f32(false, a, false, bm,
                                                      (short)0, c, false, false);
        }
#pragma unroll
        for (int r = 0; r < 8; ++r) {
            const int m = half ? (r + 8) : r;
            FIN[m * EMB + n0 + lm] = c[r];
        }
    }
    __syncthreads();

    // ---------------- Phase E: residual + L2-normalize + store -------------
    for (int rr = 0; rr < 2; ++rr) {
        const int b  = wave * 2 + rr;
        const int bg = blk * TB + b;
        const int node = g_node_i[bg];
        const int lay  = g_layers[bg];
        const float4* emb  = (const float4*)(g_layemb + ((long)node * LAYERS + lay) * EMB);
        const float4* frow = (const float4*)(FIN + b * EMB);
        float4 v = frow[lane];
        const float4 e = emb[lane];
        v.x += e.x; v.y += e.y; v.z += e.z; v.w += e.w;
        float ss = v.x*v.x + v.y*v.y + v.z*v.z + v.w*v.w;
#pragma unroll
        for (int off = 16; off > 0; off >>= 1) ss += __shfl_xor(ss, off);
        const float inv = 1.f / fmaxf(sqrtf(ss), 1e-12f);
        float4 o = { v.x * inv, v.y * inv, v.z * inv, v.w * inv };
        ((float4*)(g_out + (long)bg * EMB))[lane] = o;
    }
}

extern "C" void kernel_launch(void* const* d_in, const int* in_sizes, int n_in,
                              void* d_out, int out_size, void* d_ws, size_t ws_size,
                              hipStream_t stream) {
    const int*   layers   = (const int*)  d_in[0];
    const int*   node_i   = (const int*)  d_in[1];
    const int*   neighs   = (const int*)  d_in[2];
    const float* features = (const float*)d_in[3];
    const float* layemb   = (const float*)d_in[4];
    const float* net      = (const float*)d_in[5];
    const float* tw       = (const float*)d_in[6];
    const float* ts1      = (const float*)d_in[7];
    const float* ts2      = (const float*)d_in[8];
    float* out = (float*)d_out;

    dim3 grid(BATCH / TB);
    dim3 block(THREADS);
    liamne_fused<<<grid, block, 0, stream>>>(layers, node_i, neighs, features,
                                             layemb, net, tw, ts1, ts2, out);
}